// TemporalGNN_47931835023433
// MI455X (gfx1250) — compile-verified
//
#include <hip/hip_runtime.h>

typedef __attribute__((ext_vector_type(2))) float v2f;
typedef __attribute__((ext_vector_type(8))) float v8f;
typedef long long eidx_t;   // reference edge_index dtype is int64

#define GCN_IN  16
#define GCN_HID 64
#define GCN_OUT 32

// ---------------------------------------------------------------------------
// H = X @ W   (X: [M x K] row-major, W: [K x NOUT] row-major, H: [M x NOUT])
// One wave32 per 16x16 output tile, accumulated with V_WMMA_F32_16X16X4_F32.
// fp32 A layout (16x4): lanes 0-15 hold {K=k,k+1} of row M=lane,
//                       lanes 16-31 hold {K=k+2,k+3} of row M=lane-16.
// fp32 B layout (4x16): VGPR0 = row k (+2 for upper half-lanes), VGPR1 = +1.
// C/D layout (16x16):   VGPR v = row v (lanes 0-15) / row v+8 (lanes 16-31).
// Requires M % 16 == 0 (N = 50000 = 3125 * 16).
// ---------------------------------------------------------------------------
template <int K, int NOUT>
__global__ __launch_bounds__(256)
void gcn_gemm_wmma(const float* __restrict__ X, const float* __restrict__ W,
                   float* __restrict__ H, int numRowTiles) {
  constexpr int colTiles = NOUT / 16;
  const int wave    = blockIdx.x * (blockDim.x >> 5) + (threadIdx.x >> 5);
  const int rowTile = wave / colTiles;
  const int colTile = wave - rowTile * colTiles;
  if (rowTile >= numRowTiles) return;        // wave-uniform: EXEC all-1 at WMMA

  const int lane = threadIdx.x & 31;
  const int half = lane >> 4;                // 0: lanes 0-15, 1: lanes 16-31
  const int l16  = lane & 15;
  const int rowBase = rowTile << 4;
  const int col     = (colTile << 4) + l16;

  v8f acc = {};
#pragma unroll
  for (int k = 0; k < K; k += 4) {
    const float* ap = X + (size_t)(rowBase + l16) * K + (k + 2 * half);
    v2f a; a.x = ap[0]; a.y = ap[1];
    const float* bp = W + (size_t)(k + 2 * half) * NOUT + col;
    v2f b; b.x = bp[0]; b.y = bp[NOUT];
    // (neg_a, A, neg_b, B, c_mod, C, reuse_a, reuse_b)
    acc = __builtin_amdgcn_wmma_f32_16x16x4_f32(false, a, false, b, (short)0,
                                                acc, false, false);
  }

  float* outp = H + (size_t)(rowBase + 8 * half) * NOUT + col;
#pragma unroll
  for (int v = 0; v < 8; ++v) outp[(size_t)v * NOUT] = acc[v];
}

// ---------------------------------------------------------------------------
__global__ void gcn_degree(const eidx_t* __restrict__ dst,
                           float* __restrict__ deg, int E) {
  int e = blockIdx.x * blockDim.x + threadIdx.x;
  if (e < E) atomicAdd(&deg[(int)dst[e]], 1.0f);
}

__global__ void gcn_dinv(float* __restrict__ deg, int N) {
  int i = blockIdx.x * blockDim.x + threadIdx.x;
  if (i < N) deg[i] = rsqrtf(deg[i] + 1.0f);   // +1 self-loop => deg >= 1
}

// agg[dst] += (dinv[src]*dinv[dst]) * H[src]; D consecutive lanes per edge.
template <int D>
__global__ __launch_bounds__(256)
void gcn_scatter(const float* __restrict__ H, const eidx_t* __restrict__ src,
                 const eidx_t* __restrict__ dst, const float* __restrict__ dinv,
                 float* __restrict__ agg, int E) {
  constexpr int EPB = 256 / D;
  const int e = blockIdx.x * EPB + (int)(threadIdx.x / D);
  const int d = threadIdx.x & (D - 1);
  if (e >= E) return;
  const int s = (int)src[e];
  const int t = (int)dst[e];
  const float norm = dinv[s] * dinv[t];
  atomicAdd(&agg[(size_t)t * D + d], H[(size_t)s * D + d] * norm);
}

// agg = [relu]( agg + dinv[i]^2 * H + bias )   (self-loop fused)
template <int D, bool RELU>
__global__ __launch_bounds__(256)
void gcn_finish(float* __restrict__ agg, const float* __restrict__ H,
                const float* __restrict__ dinv, const float* __restrict__ bias,
                int N) {
  const long long idx = (long long)blockIdx.x * blockDim.x + threadIdx.x;
  const int i = (int)(idx / D);
  const int d = (int)(idx & (D - 1));
  if (i >= N) return;
  const float di = dinv[i];
  float v = agg[idx] + di * di * H[idx] + bias[d];
  agg[idx] = RELU ? fmaxf(v, 0.0f) : v;
}

// ---------------------------------------------------------------------------
extern "C" void kernel_launch(void* const* d_in, const int* in_sizes, int n_in,
                              void* d_out, int out_size, void* d_ws,
                              size_t ws_size, hipStream_t stream) {
  const float*  x   = (const float*)d_in[0];
  const eidx_t* ei  = (const eidx_t*)d_in[1];
  const float*  W1  = (const float*)d_in[2];
  const float*  b1  = (const float*)d_in[3];
  const float*  W2  = (const float*)d_in[4];
  const float*  b2  = (const float*)d_in[5];

  const int N = in_sizes[0] / GCN_IN;   // 50000
  const int E = in_sizes[1] / 2;        // 800000
  const eidx_t* src = ei;
  const eidx_t* dst = ei + E;

  // Workspace layout: dinv | h1 | agg1 | h2   (~32.2 MB total)
  float* dinv = (float*)d_ws;
  float* h1   = dinv + N;
  float* agg1 = h1 + (size_t)N * GCN_HID;
  float* h2   = agg1 + (size_t)N * GCN_HID;
  float* out  = (float*)d_out;

  hipMemsetAsync(dinv, 0, sizeof(float) * (size_t)N, stream);
  hipMemsetAsync(agg1, 0, sizeof(float) * (size_t)N * GCN_HID, stream);
  hipMemsetAsync(out,  0, sizeof(float) * (size_t)N * GCN_OUT, stream);

  gcn_degree<<<(E + 255) / 256, 256, 0, stream>>>(dst, dinv, E);
  gcn_dinv<<<(N + 255) / 256, 256, 0, stream>>>(dinv, N);

  const int rowTiles = N / 16;          // 3125, exact

  // ---- layer 1: h1 = x @ W1 ; agg1 = scatter ; relu(+self-loop,+b1) -------
  {
    const int waves = rowTiles * (GCN_HID / 16);
    gcn_gemm_wmma<GCN_IN, GCN_HID>
        <<<(waves + 7) / 8, 256, 0, stream>>>(x, W1, h1, rowTiles);
  }
  gcn_scatter<GCN_HID>
      <<<(E + (256 / GCN_HID) - 1) / (256 / GCN_HID), 256, 0, stream>>>(
          h1, src, dst, dinv, agg1, E);
  {
    const long long tot = (long long)N * GCN_HID;
    gcn_finish<GCN_HID, true>
        <<<(unsigned)((tot + 255) / 256), 256, 0, stream>>>(agg1, h1, dinv, b1, N);
  }

  // ---- layer 2: h2 = agg1 @ W2 ; out = scatter ; +self-loop,+b2 -----------
  {
    const int waves = rowTiles * (GCN_OUT / 16);
    gcn_gemm_wmma<GCN_HID, GCN_OUT>
        <<<(waves + 7) / 8, 256, 0, stream>>>(agg1, W2, h2, rowTiles);
  }
  gcn_scatter<GCN_OUT>
      <<<(E + (256 / GCN_OUT) - 1) / (256 / GCN_OUT), 256, 0, stream>>>(
          h2, src, dst, dinv, out, E);
  {
    const long long tot = (long long)N * GCN_OUT;
    gcn_finish<GCN_OUT, false>
        <<<(unsigned)((tot + 255) / 256), 256, 0, stream>>>(out, h2, dinv, b2, N);
  }
}